// MultiHeadAttention_76639396430114
// MI455X (gfx1250) — compile-verified
//
#include <hip/hip_runtime.h>

typedef __attribute__((ext_vector_type(2))) float v2f;
typedef __attribute__((ext_vector_type(8))) float v8f;

namespace {
constexpr int   kN  = 61;
constexpr int   kE  = 40;
constexpr int   kH  = 10;
constexpr int   kD  = 4;
constexpr int   kRS = 44;   // LDS row stride (floats): 44*i mod 64 distinct for i<16,
                            // and 176B rows keep float4 rows 16B-aligned
constexpr int   kRows = 64; // padded rows (61 -> 4 tiles of 16)
constexpr int   kDump = kRows * kRS;      // scratch slot for discarded stores
constexpr int   kBuf  = kRows * kRS + 16; // keeps successive buffers 16B-aligned
constexpr float kInvSqrtE = 0.15811388300841898f; // 1/sqrt(40)
}

__device__ __forceinline__ int clampi(int v, int hi2) { return v < hi2 ? v : hi2; }

__global__ __launch_bounds__(256)
void mha_fused_kernel(const float* __restrict__ x,
                      const float* __restrict__ Wq, const float* __restrict__ bq,
                      const float* __restrict__ Wk, const float* __restrict__ bk,
                      const float* __restrict__ Wv, const float* __restrict__ bv,
                      const float* __restrict__ Wp, const float* __restrict__ bp,
                      const float* __restrict__ pe,
                      float* __restrict__ out) {
  __shared__ __align__(16) float s_x[kBuf];  // x + pe; later aliased as attention out O
  __shared__ __align__(16) float s_q[kBuf];
  __shared__ __align__(16) float s_k[kBuf];
  __shared__ __align__(16) float s_v[kBuf];

  const int b    = blockIdx.x;
  const int tid  = threadIdx.x;
  const int lane = tid & 31;
  const int wave = tid >> 5;
  const int hi   = lane >> 4;   // which 16-lane half of the wave
  const int l16  = lane & 15;
  const int kbl  = hi * 2;      // K sub-offset inside a 16x16x4 A/B fragment

  if (tid == 0) {  // warm weight cachelines (global_prefetch_b8)
    __builtin_prefetch(Wq, 0, 1);
    __builtin_prefetch(Wk, 0, 1);
    __builtin_prefetch(Wv, 0, 1);
    __builtin_prefetch(Wp, 0, 1);
  }

  // ---------- Phase 1: xpe = x[b] + pe ----------
  for (int i = tid; i < kN * kE; i += 256) {
    const int n = i / kE;
    s_x[n * kRS + (i - n * kE)] = x[(size_t)b * (kN * kE) + i] + pe[i];
  }
  __syncthreads();

  // ---------- Phase 2: Q/K/V = xpe @ W^T + bias (WMMA f32 16x16x4), tril mask on K,V ----
  // Fragment loads are UNCONDITIONAL with clamped indices: garbage in padded A-rows /
  // B-cols only corrupts padded D rows/cols. Stores are UNCONDITIONAL too: invalid
  // elements are redirected to a dump slot (cndmask on the address, no EXEC juggling).
  for (int tile = wave; tile < 36; tile += 8) {
    const int mat = tile / 12;             // 0=Q 1=K 2=V
    const int mt  = (tile % 12) / 3;
    const int nt  = tile % 3;
    const float* __restrict__ W  = (mat == 0) ? Wq : (mat == 1) ? Wk : Wv;
    const float* __restrict__ bi = (mat == 0) ? bq : (mat == 1) ? bk : bv;
    float* __restrict__ S        = (mat == 0) ? s_q : (mat == 1) ? s_k : s_v;

    const int arow = mt * 16 + l16;
    const int ncol = nt * 16 + l16;
    const int ar_c = clampi(arow, kN - 1);
    const int nc_c = clampi(ncol, kE - 1);
    const bool cok = ncol < kE;

    v2f af[10], bf[10];
#pragma unroll
    for (int kc = 0; kc < 10; ++kc) {
      const int kb = kc * 4 + kbl;
      af[kc].x = s_x[ar_c * kRS + kb];
      af[kc].y = s_x[ar_c * kRS + kb + 1];
      bf[kc].x = W[nc_c * kE + kb];        // B[k][n] = W[n][k]
      bf[kc].y = W[nc_c * kE + kb + 1];
    }
    v8f acc = {0.f, 0.f, 0.f, 0.f, 0.f, 0.f, 0.f, 0.f};
#pragma unroll
    for (int kc = 0; kc < 10; ++kc)
      acc = __builtin_amdgcn_wmma_f32_16x16x4_f32(false, af[kc], false, bf[kc],
                                                  (short)0, acc, false, false);

    const float bias = bi[nc_c];
#pragma unroll
    for (int r = 0; r < 8; ++r) {
      const int row = mt * 16 + r + hi * 8;
      float val = acc[r] + bias;
      // tril(ones(n,D)) mask on K,V: (ncol&3) > row can only fire for row <= 2,
      // i.e. mt==0, hi==0, r<3 -- compile-time pruned to 3 selects per tile.
      if (r < 3)
        val = (mat != 0 && mt == 0 && hi == 0 && (ncol & 3) > r) ? 0.f : val;
      const bool ok = (row < kN) && cok;
      S[ok ? (row * kRS + ncol) : kDump] = val;   // unconditional store
    }
  }
  __syncthreads();

  float* __restrict__ s_o = s_x;  // alias: xpe dead after projections

  // ---------- Phase 3: attention, one head per wave ----------
  // Energy is computed TRANSPOSED: E^T = K @ Q^T  (A = K rows=keys, B = Q^T cols=queries).
  // C layout then gives each lane 32 of the 61 key scores of ITS OWN query in registers:
  // softmax reduces locally + one shfl_xor(16); att@V is register FMAs against
  // half-uniform (broadcast) V rows; each lane owns all 4 output dims of its query.
  for (int h = wave; h < kH; h += 8) {
    const int co = h * kD;

    v2f ak[4];                                    // A fragments: K rows (keys)
#pragma unroll
    for (int kt = 0; kt < 4; ++kt) {
      const int kr = clampi(kt * 16 + l16, kN - 1);
      ak[kt].x = s_k[kr * kRS + co + kbl];
      ak[kt].y = s_k[kr * kRS + co + kbl + 1];
    }

#pragma unroll
    for (int qt = 0; qt < 4; ++qt) {
      const int q   = qt * 16 + l16;              // this lane's query row
      const int q_c = clampi(q, kN - 1);
      v2f bq_;                                    // B fragment: Q^T column = query
      bq_.x = s_q[q_c * kRS + co + kbl];
      bq_.y = s_q[q_c * kRS + co + kbl + 1];

      v8f et[4];
#pragma unroll
      for (int kt = 0; kt < 4; ++kt) {
        v8f z = {0.f, 0.f, 0.f, 0.f, 0.f, 0.f, 0.f, 0.f};
        et[kt] = __builtin_amdgcn_wmma_f32_16x16x4_f32(false, ak[kt], false, bq_,
                                                       (short)0, z, false, false);
      }
      // invalid keys 61..63 live at kt==3, r=5..7, hi==1
#pragma unroll
      for (int r = 5; r < 8; ++r) et[3][r] = hi ? -1e30f : et[3][r];

      // ---- softmax over keys (register-local + one cross-half shuffle) ----
      float m = -1e30f;
#pragma unroll
      for (int kt = 0; kt < 4; ++kt)
#pragma unroll
        for (int r = 0; r < 8; ++r) m = fmaxf(m, et[kt][r]);
      m = fmaxf(m, __shfl_xor(m, 16, 32));

      float sum = 0.f;
#pragma unroll
      for (int kt = 0; kt < 4; ++kt)
#pragma unroll
        for (int r = 0; r < 8; ++r) {
          const float e = __expf((et[kt][r] - m) * kInvSqrtE);
          et[kt][r] = e;
          sum += e;
        }
      sum += __shfl_xor(sum, 16, 32);
      const float inv = 1.f / sum;

      // ---- out = att @ V : FMAs against half-uniform V rows (LDS broadcast) ----
      float o0 = 0.f, o1 = 0.f, o2 = 0.f, o3 = 0.f;
#pragma unroll
      for (int kt = 0; kt < 4; ++kt)
#pragma unroll
        for (int r = 0; r < 8; ++r) {
          const int key = clampi(kt * 16 + r + 8 * hi, kN - 1);
          const float4 vr = *(const float4*)(s_v + key * kRS + co);  // 16B-aligned
          const float a = et[kt][r];
          o0 += a * vr.x; o1 += a * vr.y; o2 += a * vr.z; o3 += a * vr.w;
        }
      o0 += __shfl_xor(o0, 16, 32);
      o1 += __shfl_xor(o1, 16, 32);
      o2 += __shfl_xor(o2, 16, 32);
      o3 += __shfl_xor(o3, 16, 32);

      float4 ov; ov.x = o0 * inv; ov.y = o1 * inv; ov.z = o2 * inv; ov.w = o3 * inv;
      // unconditional float4 store; invalid queries land in the (16B-aligned) dump slot
      *(float4*)(s_o + (q < kN ? q * kRS + co : kDump)) = ov;
    }
  }
  __syncthreads();

  // ---------- Phase 4: out = O @ Wp^T + bp -> global ----------
  for (int tile = wave; tile < 12; tile += 8) {
    const int mt   = tile / 3;
    const int nt   = tile % 3;
    const int arow = mt * 16 + l16;
    const int ncol = nt * 16 + l16;
    const int ar_c = clampi(arow, kN - 1);
    const int nc_c = clampi(ncol, kE - 1);
    const bool cok = ncol < kE;

    v2f af[10], bf[10];
#pragma unroll
    for (int kc = 0; kc < 10; ++kc) {
      const int kb = kc * 4 + kbl;
      af[kc].x = s_o[ar_c * kRS + kb];
      af[kc].y = s_o[ar_c * kRS + kb + 1];
      bf[kc].x = Wp[nc_c * kE + kb];
      bf[kc].y = Wp[nc_c * kE + kb + 1];
    }
    v8f acc = {0.f, 0.f, 0.f, 0.f, 0.f, 0.f, 0.f, 0.f};
#pragma unroll
    for (int kc = 0; kc < 10; ++kc)
      acc = __builtin_amdgcn_wmma_f32_16x16x4_f32(false, af[kc], false, bf[kc],
                                                  (short)0, acc, false, false);

    const float bias = bp[nc_c];
#pragma unroll
    for (int r = 0; r < 8; ++r) {
      const int row = mt * 16 + r + hi * 8;
      if (row < kN && cok)
        out[(size_t)b * (kN * kE) + row * kE + ncol] = acc[r] + bias;
    }
  }
}

extern "C" void kernel_launch(void* const* d_in, const int* in_sizes, int n_in,
                              void* d_out, int out_size, void* d_ws, size_t ws_size,
                              hipStream_t stream) {
  (void)in_sizes; (void)n_in; (void)out_size; (void)d_ws; (void)ws_size;
  const float* x  = (const float*)d_in[0];
  const float* Wq = (const float*)d_in[1];
  const float* bq = (const float*)d_in[2];
  const float* Wk = (const float*)d_in[3];
  const float* bk = (const float*)d_in[4];
  const float* Wv = (const float*)d_in[5];
  const float* bv = (const float*)d_in[6];
  const float* Wp = (const float*)d_in[7];
  const float* bp = (const float*)d_in[8];
  const float* pe = (const float*)d_in[9];
  float* out = (float*)d_out;

  mha_fused_kernel<<<dim3(4096), dim3(256), 0, stream>>>(
      x, Wq, bq, Wk, bk, Wv, bv, Wp, bp, pe, out);
}